// LISTA_21595095564710
// MI455X (gfx1250) — compile-verified
//
#include <hip/hip_runtime.h>

// LISTA on MI455X (gfx1250): single persistent workgroup, 4 waves (wave32).
// Complex 8x8 matmul folded to real 16x16 via [[Wr,-Wi],[Wi,Wr]]; each wave
// owns one 16-column tile of the 16x64 state and runs V_WMMA_F32_16X16X4_F32
// (K accumulated in 4 chunks). WMMA D-layout leaves each lane holding a full
// length-8 z column (z_r for lanes 0-15, z_i for lanes 16-31), so the conv
// pipeline is per-lane VALU. LDS is only an intra-wave layout shuffle.

typedef __attribute__((ext_vector_type(2))) float v2f;
typedef __attribute__((ext_vector_type(8))) float v8f;

#define STAGES 16
#define LDS_STRIDE 68  // 16 rows x 68: breaks bank conflicts between half-waves

__device__ __forceinline__ float soft_comp(float x, float t) {
  return x > t ? x - t : (x < -t ? x + t : 0.0f);
}

// Wbig = [[wr, -wi],[wi, wr]], each block 8x8 row-major (one stage slice).
__device__ __forceinline__ float wbig_load(const float* __restrict__ wr,
                                           const float* __restrict__ wi,
                                           int r, int c) {
  const int rr = r & 7, cc = c & 7;
  const bool rlo = (r < 8), clo = (c < 8);
  const float* p = (rlo == clo) ? wr : wi;
  float v = p[rr * 8 + cc];
  return (rlo && !clo) ? -v : v;
}

__launch_bounds__(128, 1)
__global__ void lista_cdna5_kernel(const float* __restrict__ y_real,
                                   const float* __restrict__ y_imag,
                                   const float* __restrict__ w1_real,
                                   const float* __restrict__ w1_imag,
                                   const float* __restrict__ w2_real,
                                   const float* __restrict__ w2_imag,
                                   const float* __restrict__ thr,
                                   const float* __restrict__ conv1_w,
                                   const float* __restrict__ conv1_b,
                                   const float* __restrict__ conv2_w,
                                   const float* __restrict__ conv2_b,
                                   float* __restrict__ out) {
  __shared__ float Xlds[16 * LDS_STRIDE];

  const int tid   = threadIdx.x;
  const int wave  = tid >> 5;        // 0..3 : 16-column tile
  const int lane  = tid & 31;
  const int khalf = lane >> 4;       // 0 or 1 (half-wave)
  const int l16   = lane & 15;
  const int row   = l16;             // A-matrix row for this lane
  const int col   = wave * 16 + l16; // global column 0..63

  // Y fragments in B-matrix layout (stage-invariant, kept in registers).
  // B-frag chunk k, vgpr v holds Ybig[4k + v + 2*khalf][col].
  float yfrag[4][2];
#pragma unroll
  for (int k = 0; k < 4; ++k) {
#pragma unroll
    for (int v = 0; v < 2; ++v) {
      int kk = 4 * k + v + 2 * khalf;
      yfrag[k][v] = (kk < 8) ? y_real[kk * 64 + col]
                             : y_imag[(kk - 8) * 64 + col];
    }
  }

  // ---- init: X0 = soft(W1big(stage 0) @ Ybig, thr[0]) ----
  {
    v8f acc = {};
#pragma unroll
    for (int k = 0; k < 4; ++k) {
      v2f a, b;
      a.x = wbig_load(w1_real, w1_imag, row, 4 * k + 0 + 2 * khalf);
      a.y = wbig_load(w1_real, w1_imag, row, 4 * k + 1 + 2 * khalf);
      b.x = yfrag[k][0];
      b.y = yfrag[k][1];
      acc = __builtin_amdgcn_wmma_f32_16x16x4_f32(false, a, false, b,
                                                  (short)0, acc, false, false);
    }
    float t0 = thr[0];
#pragma unroll
    for (int j = 0; j < 8; ++j)
      Xlds[(j + 8 * khalf) * LDS_STRIDE + col] = soft_comp(acc[j], t0);
  }
  __syncthreads();

  float x[8];
  for (int s = 0; s < STAGES; ++s) {
    const float* w2r = w2_real + s * 64;
    const float* w2i = w2_imag + s * 64;
    const float* w1r = w1_real + s * 64;
    const float* w1i = w1_imag + s * 64;

    v8f acc = {};
    // Z += W2big @ X   (X fragments pulled from LDS in B layout)
#pragma unroll
    for (int k = 0; k < 4; ++k) {
      v2f a, b;
      a.x = wbig_load(w2r, w2i, row, 4 * k + 0 + 2 * khalf);
      a.y = wbig_load(w2r, w2i, row, 4 * k + 1 + 2 * khalf);
      int kk = 4 * k + 2 * khalf;
      b.x = Xlds[(kk + 0) * LDS_STRIDE + col];
      b.y = Xlds[(kk + 1) * LDS_STRIDE + col];
      acc = __builtin_amdgcn_wmma_f32_16x16x4_f32(false, a, false, b,
                                                  (short)0, acc, false, false);
    }
    // Z += W1big @ Y   (Y fragments from registers)
#pragma unroll
    for (int k = 0; k < 4; ++k) {
      v2f a, b;
      a.x = wbig_load(w1r, w1i, row, 4 * k + 0 + 2 * khalf);
      a.y = wbig_load(w1r, w1i, row, 4 * k + 1 + 2 * khalf);
      b.x = yfrag[k][0];
      b.y = yfrag[k][1];
      acc = __builtin_amdgcn_wmma_f32_16x16x4_f32(false, a, false, b,
                                                  (short)0, acc, false, false);
    }

    // acc[0..7]: full length-8 column of z_r (khalf==0) or z_i (khalf==1).
    // Tiny conv pipeline, per lane, uniform weights (scalar loads).
    float t = thr[s];
    float c1w[3][3], c1b[3], c2w[3][3];
#pragma unroll
    for (int ch = 0; ch < 3; ++ch) {
      c1b[ch] = conv1_b[s * 3 + ch];
#pragma unroll
      for (int k = 0; k < 3; ++k) {
        c1w[ch][k] = conv1_w[s * 9 + ch * 3 + k];  // (16,3,1,3)
        c2w[ch][k] = conv2_w[s * 9 + ch * 3 + k];  // (16,1,3,3)
      }
    }
    float b2 = conv2_b[s];

#pragma unroll
    for (int j = 0; j < 8; ++j) x[j] = b2;
#pragma unroll
    for (int ch = 0; ch < 3; ++ch) {
      float u[8];
#pragma unroll
      for (int j = 0; j < 8; ++j) {
        float zm = (j > 0) ? acc[j - 1] : 0.0f;
        float zp = (j < 7) ? acc[j + 1] : 0.0f;
        float uu = c1b[ch] + c1w[ch][0] * zm + c1w[ch][1] * acc[j]
                 + c1w[ch][2] * zp;
        u[j] = soft_comp(uu, t);
      }
#pragma unroll
      for (int j = 0; j < 8; ++j) {
        float um = (j > 0) ? u[j - 1] : 0.0f;
        float up = (j < 7) ? u[j + 1] : 0.0f;
        x[j] += c2w[ch][0] * um + c2w[ch][1] * u[j] + c2w[ch][2] * up;
      }
    }

    __syncthreads();  // intra-wave only in principle; cheap ordering insurance
#pragma unroll
    for (int j = 0; j < 8; ++j)
      Xlds[(j + 8 * khalf) * LDS_STRIDE + col] = x[j];
    __syncthreads();
  }

  // Output: x_r (8x64) then x_i (8x64), row-major, straight from registers.
  float* o = out + khalf * 512;
#pragma unroll
  for (int j = 0; j < 8; ++j) o[j * 64 + col] = x[j];
}

extern "C" void kernel_launch(void* const* d_in, const int* in_sizes, int n_in,
                              void* d_out, int out_size, void* d_ws, size_t ws_size,
                              hipStream_t stream) {
  const float* y_real  = (const float*)d_in[0];
  const float* y_imag  = (const float*)d_in[1];
  const float* w1_real = (const float*)d_in[2];
  const float* w1_imag = (const float*)d_in[3];
  const float* w2_real = (const float*)d_in[4];
  const float* w2_imag = (const float*)d_in[5];
  const float* thr     = (const float*)d_in[6];
  const float* conv1_w = (const float*)d_in[7];
  const float* conv1_b = (const float*)d_in[8];
  const float* conv2_w = (const float*)d_in[9];
  const float* conv2_b = (const float*)d_in[10];

  lista_cdna5_kernel<<<1, 128, 0, stream>>>(
      y_real, y_imag, w1_real, w1_imag, w2_real, w2_imag, thr,
      conv1_w, conv1_b, conv2_w, conv2_b, (float*)d_out);
}